// ModelCrf4Obj_91113436217406
// MI455X (gfx1250) — compile-verified
//
#include <hip/hip_runtime.h>
#include <hip/hip_bf16.h>

// Problem constants (B=32, S=512, D=768, K=3)
#define BB 32
#define SS 512
#define DD 768
#define KK 3
#define ROWS (BB * SS)          // 16384
#define DD2 (2 * DD)            // 1536
#define NIT (DD2 / 4)           // 384 WMMA iterations along the reduction

typedef __attribute__((ext_vector_type(2))) float v2f;
typedef __attribute__((ext_vector_type(8))) float v8f;

// ---------------------------------------------------------------------------
// Kernel 1: build the WMMA B-fragment table directly.
//   W_eff[k][n] = sum_f W_cnn[k][f] * W_dense[f][n]   (k in [0,1536), n in [0,3))
// Table entry (it, lane), it = k0/4, lane in [0,32):
//   half = lane>>4, n = lane&15, k = it*4 + 2*half
//   Btab[it*32 + lane] = { W_eff[k][n], W_eff[k+1][n] }   (0 for n >= 3)
// Each nonzero W_eff element lands in exactly one table slot -> no redundancy.
// Also: b_eff = b_cnn @ W_dense + b_dense.
// ---------------------------------------------------------------------------
__global__ void weff_frag_kernel(const float* __restrict__ Wcnn,
                                 const float* __restrict__ bcnn,
                                 const float* __restrict__ Wdense,
                                 const float* __restrict__ bdense,
                                 v2f* __restrict__ Btab,
                                 float* __restrict__ beff) {
    int id = blockIdx.x * blockDim.x + threadIdx.x;
    if (id < NIT * 32) {
        int lane = id & 31;
        int it   = id >> 5;
        int half = lane >> 4;
        int n    = lane & 15;
        int k    = it * 4 + 2 * half;
        float w0 = 0.f, w1 = 0.f;
        if (n < KK) {
            const float* row0 = Wcnn + (size_t)k * DD;
            const float* row1 = row0 + DD;
            for (int f = 0; f < DD; ++f) {
                float wd = Wdense[f * KK + n];
                w0 += row0[f] * wd;
                w1 += row1[f] * wd;
            }
        }
        v2f b; b.x = w0; b.y = w1;
        Btab[id] = b;
    } else if (id < NIT * 32 + KK) {
        int n = id - NIT * 32;
        float acc = bdense[n];
        for (int f = 0; f < DD; ++f) acc += bcnn[f] * Wdense[f * KK + n];
        beff[n] = acc;
    }
}

// ---------------------------------------------------------------------------
// Kernel 2: logits = x @ W_eff + b_eff via V_WMMA_F32_16X16X4_F32.
// One wave per 16-row tile; 384 WMMAs (768 k-steps from `hidden`, 768 from
// `weights_sub`). Hot loop = one 8B A load (lane's row slice) + one fully
// coalesced 8B B load (precomputed fragment) + one WMMA.
//
// f32 WMMA 16x16x4 layouts (ISA 7.12.2):
//   A (16x4): lane m (0..15) row m; halves hold K = {2h, 2h+1} in 2 VGPRs.
//   B (4x16): lane col n = lane&15, same K mapping per half.
//   C/D: VGPR j -> row j + 8*half, col = lane&15.
// ---------------------------------------------------------------------------
__global__ void __launch_bounds__(256)
logits_wmma_kernel(const float* __restrict__ hidden,
                   const float* __restrict__ wsub,
                   const v2f* __restrict__ Btab,
                   const float* __restrict__ beff,
                   float* __restrict__ logits) {
    int wave = (blockIdx.x * blockDim.x + threadIdx.x) >> 5;  // 0..1023
    int lane = threadIdx.x & 31;
    int half = lane >> 4;      // 0 or 1
    int mn   = lane & 15;      // A row within tile / B,C column
    int base = wave * 16;      // first global row of this tile

    const float* rowH = hidden + (size_t)(base + mn) * DD + 2 * half;
    const float* rowW = wsub   + (size_t)(base + mn) * DD + 2 * half;
    const v2f*   bt   = Btab + lane;

    v8f c = {0.f, 0.f, 0.f, 0.f, 0.f, 0.f, 0.f, 0.f};

    // First half of the reduction: x[:, 0:768] = hidden
    #pragma unroll 4
    for (int it = 0; it < NIT / 2; ++it) {
        v2f a = *(const v2f*)(rowH + it * 4);   // 8B aligned
        v2f b = bt[it * 32];                    // coalesced 256B per wave
        c = __builtin_amdgcn_wmma_f32_16x16x4_f32(false, a, false, b,
                                                  (short)0, c, false, false);
    }
    // Second half: x[:, 768:1536] = weights_sub
    #pragma unroll 4
    for (int it = NIT / 2; it < NIT; ++it) {
        v2f a = *(const v2f*)(rowW + (it - NIT / 2) * 4);
        v2f b = bt[it * 32];
        c = __builtin_amdgcn_wmma_f32_16x16x4_f32(false, a, false, b,
                                                  (short)0, c, false, false);
    }

    if (mn < KK) {
        float bias = beff[mn];
        #pragma unroll
        for (int j = 0; j < 8; ++j) {
            int row = base + j + 8 * half;
            logits[(size_t)row * KK + mn] = c[j] + bias;
        }
    }
}

// ---------------------------------------------------------------------------
// Kernel 3: CRF (forward logsumexp + score + Viterbi + backtrace).
// One 128-thread block; 4 lanes per batch (j = lane%4, lanes j<3 live).
// Cross-lane alpha exchange via wave32 __shfl; Viterbi backpointers in LDS.
// Outputs: out[0 .. 16383] = tags (as float), out[16384 .. 16415] = log-lik.
// ---------------------------------------------------------------------------
__global__ void __launch_bounds__(128)
crf_kernel(const float* __restrict__ logits,
           const float* __restrict__ trans,
           const int* __restrict__ obj,
           float* __restrict__ out) {
    __shared__ unsigned char bp[BB][SS][KK];   // 48 KB backpointers

    int tid = threadIdx.x;        // 0..127
    int b   = tid >> 2;           // batch 0..31
    int j   = tid & 3;            // 0..3 (j==3 is a mirror lane)
    int jj  = (j < 3) ? j : 2;
    int lw  = tid & 31;           // lane within wave
    int gb  = lw & ~3;            // group base lane

    float tr[9];
    #pragma unroll
    for (int i = 0; i < 9; ++i) tr[i] = trans[i];

    const int*   ob = obj + b * SS;
    const float* lg = logits + (size_t)b * SS * KK;

    // seq_len = count(obj_target >= 1)
    int cnt = 0;
    for (int s = j; s < SS; s += 4) cnt += (ob[s] >= 1) ? 1 : 0;
    int L = __shfl(cnt, gb + 0, 32) + __shfl(cnt, gb + 1, 32) +
            __shfl(cnt, gb + 2, 32) + __shfl(cnt, gb + 3, 32);
    int idx = L - 1;
    if (idx < 0) idx = 0;
    if (idx > SS - 1) idx = SS - 1;

    float alpha     = lg[jj];     // forward alpha (unmasked scan, as in ref)
    float valpha    = alpha;      // viterbi alpha (masked updates)
    float alpha_idx = alpha;      // alpha at t == idx
    if (j < 3) bp[b][0][jj] = 0;  // bp_full[0] = 0

    for (int t = 1; t < SS; ++t) {
        float pot = lg[t * KK + jj];

        // forward: alpha_t[j] = pot + logsumexp_i(alpha[i] + trans[i][j])
        float a0 = __shfl(alpha, gb + 0, 32);
        float a1 = __shfl(alpha, gb + 1, 32);
        float a2 = __shfl(alpha, gb + 2, 32);
        float s0 = a0 + tr[0 * KK + jj];
        float s1 = a1 + tr[1 * KK + jj];
        float s2 = a2 + tr[2 * KK + jj];
        float m  = fmaxf(s0, fmaxf(s1, s2));
        float sm = expf(s0 - m) + expf(s1 - m) + expf(s2 - m);
        alpha = pot + m + logf(sm);
        if (t == idx) alpha_idx = alpha;

        // viterbi (first-occurrence argmax, masked update)
        float v0 = __shfl(valpha, gb + 0, 32);
        float v1 = __shfl(valpha, gb + 1, 32);
        float v2 = __shfl(valpha, gb + 2, 32);
        float w0 = v0 + tr[0 * KK + jj];
        float w1 = v1 + tr[1 * KK + jj];
        float w2 = v2 + tr[2 * KK + jj];
        float vm = w0; int arg = 0;
        if (w1 > vm) { vm = w1; arg = 1; }
        if (w2 > vm) { vm = w2; arg = 2; }
        bool maskt = (t < L);
        valpha = maskt ? (pot + vm) : valpha;
        int bpv = maskt ? arg : 0;
        if (j < 3) bp[b][t][jj] = (unsigned char)bpv;
    }

    // log_norm = logsumexp(alpha at idx); 0 if seq_len <= 0
    float b0 = __shfl(alpha_idx, gb + 0, 32);
    float b1 = __shfl(alpha_idx, gb + 1, 32);
    float b2 = __shfl(alpha_idx, gb + 2, 32);
    float mm = fmaxf(b0, fmaxf(b1, b2));
    float ln = mm + logf(expf(b0 - mm) + expf(b1 - mm) + expf(b2 - mm));
    if (L <= 0) ln = 0.f;

    // init_tag = argmax_j final viterbi alpha (first occurrence)
    float f0 = __shfl(valpha, gb + 0, 32);
    float f1 = __shfl(valpha, gb + 1, 32);
    float f2 = __shfl(valpha, gb + 2, 32);
    int init_tag = 0; float fm = f0;
    if (f1 > fm) { fm = f1; init_tag = 1; }
    if (f2 > fm) { fm = f2; init_tag = 2; }

    // unary + binary scores (4-lane strided partials)
    float up = 0.f, bn = 0.f;
    for (int s = j; s < SS; s += 4) {
        int tg = ob[s];
        if (s < L) up += lg[s * KK + tg];
        if (s < SS - 1 && s < L - 1) bn += tr[tg * KK + ob[s + 1]];
    }
    float unary  = __shfl(up, gb + 0, 32) + __shfl(up, gb + 1, 32) +
                   __shfl(up, gb + 2, 32) + __shfl(up, gb + 3, 32);
    float binary = __shfl(bn, gb + 0, 32) + __shfl(bn, gb + 1, 32) +
                   __shfl(bn, gb + 2, 32) + __shfl(bn, gb + 3, 32);

    __syncthreads();  // LDS backpointers fully visible

    if (j == 0) {
        out[ROWS + b] = unary + binary - ln;
        // backtrace: tags[t] = 0 for t>L-1, init_tag at t==L-1, bp below
        int prev = 0;
        int Lm1 = L - 1;
        for (int t = SS - 1; t >= 0; --t) {
            int tn = (t + 1 > SS - 1) ? (SS - 1) : (t + 1);
            int fb = (int)bp[b][tn][prev];
            int tag = (t == Lm1) ? init_tag : ((t < Lm1) ? fb : 0);
            out[b * SS + t] = (float)tag;
            prev = tag;
        }
    }
}

// ---------------------------------------------------------------------------
extern "C" void kernel_launch(void* const* d_in, const int* in_sizes, int n_in,
                              void* d_out, int out_size, void* d_ws, size_t ws_size,
                              hipStream_t stream) {
    const float* hidden = (const float*)d_in[0];
    const float* wsub   = (const float*)d_in[1];
    const float* Wcnn   = (const float*)d_in[2];
    const float* bcnn   = (const float*)d_in[3];
    const float* Wdense = (const float*)d_in[4];
    const float* bdense = (const float*)d_in[5];
    const float* trans  = (const float*)d_in[6];
    const int*   obj    = (const int*)d_in[7];
    float* out = (float*)d_out;

    float* ws     = (float*)d_ws;
    v2f*   Btab   = (v2f*)ws;            // 384*32 v2f = 24576 floats (96 KB)
    float* beff   = ws + NIT * 64;       // 3 floats (16 reserved)
    float* logits = ws + NIT * 64 + 16;  // 16384*3 = 49152 floats
    // total workspace: ~295 KB

    weff_frag_kernel<<<49, 256, 0, stream>>>(Wcnn, bcnn, Wdense, bdense, Btab, beff);
    logits_wmma_kernel<<<128, 256, 0, stream>>>(hidden, wsub, Btab, beff, logits);
    crf_kernel<<<1, 128, 0, stream>>>(logits, trans, obj, out);
}